// GCN_40922448396696
// MI455X (gfx1250) — compile-verified
//
#include <hip/hip_runtime.h>

typedef __attribute__((ext_vector_type(2))) float v2f;
typedef __attribute__((ext_vector_type(8))) float v8f;

#define IN_DIM 500
#define HID    16
#define OUTD   500

// ---------------------------------------------------------------------------
// Degree / normalization
// ---------------------------------------------------------------------------
__global__ void init_deg_kernel(float* __restrict__ deg, int n) {
  int i = blockIdx.x * blockDim.x + threadIdx.x;
  if (i < n) deg[i] = 1.0f;                       // self-loop contributes 1
}

__global__ void deg_edge_kernel(const int* __restrict__ dst, float* __restrict__ deg, int e) {
  int i = blockIdx.x * blockDim.x + threadIdx.x;
  if (i < e) atomicAdd(&deg[dst[i]], 1.0f);
}

__global__ void dis_kernel(float* __restrict__ deg, int n) {
  int i = blockIdx.x * blockDim.x + threadIdx.x;
  if (i < n) {
    float d = deg[i];
    deg[i] = (d > 0.0f) ? rsqrtf(d) : 0.0f;       // in-place: deg -> d^-1/2
  }
}

// ---------------------------------------------------------------------------
// GEMM1: H[M x 16] = X[M x 500] @ W[500 x 16]    (one wave per 16-row tile)
// f32 WMMA 16x16x4, software-pipelined: loads for step k+4 issue before the
// WMMA for step k so the wave keeps >=1 load in flight across the HBM stream.
// ---------------------------------------------------------------------------
__global__ void gemm_xw1_kernel(const float* __restrict__ X,
                                const float* __restrict__ W,
                                float* __restrict__ H, int M) {
  int tid  = blockIdx.x * blockDim.x + threadIdx.x;
  int wave = tid >> 5;
  int lane = tid & 31;
  int nTiles = M >> 4;                            // M = 100000 -> 6250 tiles
  if (wave >= nTiles) return;                     // wave-uniform exit

  int half = lane >> 4;                           // 0 | 1
  int l15  = lane & 15;
  int kOfs = half * 2;                            // A/B K offset: 0 or 2
  const float* __restrict__ xrow = X + (size_t)((wave << 4) + l15) * IN_DIM;

  // prologue: load k = 0 chunk
  v2f a0, b0;
  {
    int kk = kOfs;
    a0.x = xrow[kk];
    a0.y = xrow[kk + 1];
    b0.x = W[(size_t)kk * HID + l15];
    b0.y = W[(size_t)(kk + 1) * HID + l15];
  }
  __builtin_prefetch(xrow + 64, 0, 1);

  v8f acc = {};
  for (int k = 4; k < IN_DIM; k += 4) {
    if ((k & 31) == 0)
      __builtin_prefetch(xrow + k + 64, 0, 1);    // global_prefetch 256B ahead
    // issue next chunk's loads BEFORE consuming the current one
    v2f a1, b1;
    int kn = k + kOfs;
    a1.x = xrow[kn];
    a1.y = xrow[kn + 1];
    b1.x = W[(size_t)kn * HID + l15];
    b1.y = W[(size_t)(kn + 1) * HID + l15];
    acc = __builtin_amdgcn_wmma_f32_16x16x4_f32(false, a0, false, b0,
                                                (short)0, acc, false, false);
    a0 = a1;
    b0 = b1;
  }
  acc = __builtin_amdgcn_wmma_f32_16x16x4_f32(false, a0, false, b0,
                                              (short)0, acc, false, false);

  float* __restrict__ hrow = H + (size_t)(wave << 4) * HID;
  int rofs = half * 8;                            // D: VGPR v -> M = v + 8*half
#pragma unroll
  for (int v = 0; v < 8; ++v)
    hrow[(size_t)(v + rofs) * HID + l15] = acc[v];
}

// ---------------------------------------------------------------------------
// Generic K=16 GEMM: C[M x N] = A[M x 16] @ B[16 x N] (+ bias), N tiled by 16.
// Used for layer-2 (N=16, bias=null) and layer-3 (N=500, bias=b3, -> d_out).
// All 4 K-steps' loads are issued up front, then 4 chained WMMAs.
// ---------------------------------------------------------------------------
__global__ void gemm_k16_kernel(const float* __restrict__ A,
                                const float* __restrict__ B,
                                const float* __restrict__ bias,
                                float* __restrict__ C, int M, int N) {
  int tid  = blockIdx.x * blockDim.x + threadIdx.x;
  int wave = tid >> 5;
  int lane = tid & 31;
  int mTiles = M >> 4;
  int nTiles = (N + 15) >> 4;
  if (wave >= mTiles * nTiles) return;            // wave-uniform exit
  int tm = wave / nTiles;
  int tn = wave - tm * nTiles;

  int half = lane >> 4;
  int l15  = lane & 15;
  int col  = (tn << 4) + l15;
  bool colOK = (col < N);
  const float* __restrict__ arow = A + (size_t)((tm << 4) + l15) * HID;

  // issue all loads first (4 K-steps)
  v2f a[4], b[4];
#pragma unroll
  for (int s = 0; s < 4; ++s) {
    int kk = s * 4 + half * 2;
    a[s].x = arow[kk];
    a[s].y = arow[kk + 1];
    b[s].x = colOK ? B[(size_t)kk * N + col] : 0.0f;
    b[s].y = colOK ? B[(size_t)(kk + 1) * N + col] : 0.0f;
  }

  v8f acc = {};
#pragma unroll
  for (int s = 0; s < 4; ++s)
    acc = __builtin_amdgcn_wmma_f32_16x16x4_f32(false, a[s], false, b[s],
                                                (short)0, acc, false, false);

  if (colOK) {
    float bv = bias ? bias[col] : 0.0f;
    float* __restrict__ crow = C + (size_t)(tm << 4) * N;
    int rofs = half * 8;
#pragma unroll
    for (int v = 0; v < 8; ++v)
      crow[(size_t)(v + rofs) * N + col] = acc[v] + bv;
  }
}

// ---------------------------------------------------------------------------
// Aggregation in 16-dim space (gather src features, scatter-add to dst).
// Source (6.4MB) and target (6.4MB) live in the 192MB L2 -> L2-atomic bound.
// ---------------------------------------------------------------------------
__global__ void init_agg_kernel(const float* __restrict__ H, float* __restrict__ agg,
                                const float* __restrict__ dis, int n16) {
  int i = blockIdx.x * blockDim.x + threadIdx.x;
  if (i < n16) {
    float d = dis[i >> 4];
    agg[i] = d * d * H[i];                        // self-loop term, norm = dis[i]^2
  }
}

__global__ void edge_agg_kernel(const float* __restrict__ H, float* __restrict__ agg,
                                const float* __restrict__ dis,
                                const int* __restrict__ src, const int* __restrict__ dst,
                                int e) {
  int tid = blockIdx.x * blockDim.x + threadIdx.x;
  if (tid < e * HID) {
    int ei = tid >> 4;
    int d  = tid & 15;
    int s  = src[ei];
    int t  = dst[ei];
    float w = dis[s] * dis[t];
    atomicAdd(&agg[(size_t)t * HID + d], w * H[(size_t)s * HID + d]);
  }
}

__global__ void bias_relu_kernel(const float* __restrict__ in, const float* __restrict__ b,
                                 float* __restrict__ out, int n16) {
  int i = blockIdx.x * blockDim.x + threadIdx.x;
  if (i < n16) out[i] = fmaxf(in[i] + b[i & 15], 0.0f);
}

// ---------------------------------------------------------------------------
// Launcher
// ---------------------------------------------------------------------------
extern "C" void kernel_launch(void* const* d_in, const int* in_sizes, int n_in,
                              void* d_out, int out_size, void* d_ws, size_t ws_size,
                              hipStream_t stream) {
  const float* X  = (const float*)d_in[0];
  const int*   EI = (const int*)d_in[1];
  const float* W1 = (const float*)d_in[2];
  const float* b1 = (const float*)d_in[3];
  const float* W2 = (const float*)d_in[4];
  const float* b2 = (const float*)d_in[5];
  const float* W3 = (const float*)d_in[6];
  const float* b3 = (const float*)d_in[7];
  float* out = (float*)d_out;

  const int N = in_sizes[0] / IN_DIM;             // 100000
  const int E = in_sizes[1] / 2;                  // 3200000
  const int* src = EI;
  const int* dst = EI + E;

  // Workspace carve-up (256B aligned): dis[N], bufA[N*16], bufB[N*16] ~13.2MB
  char* ws = (char*)d_ws;
  size_t off = 0;
  float* dis = (float*)(ws + off); off += (((size_t)N * 4 + 255) & ~(size_t)255);
  float* bufA = (float*)(ws + off); off += (size_t)N * HID * 4;
  float* bufB = (float*)(ws + off);

  const int T = 256;
  const int n16 = N * HID;
  dim3 blk(T);
  auto cdiv = [](long long a, long long b) { return (unsigned)((a + b - 1) / b); };

  // normalization
  init_deg_kernel<<<cdiv(N, T), blk, 0, stream>>>(dis, N);
  deg_edge_kernel<<<cdiv(E, T), blk, 0, stream>>>(dst, dis, E);
  dis_kernel<<<cdiv(N, T), blk, 0, stream>>>(dis, N);

  const int mTiles = N >> 4;                      // 6250
  // ---- layer 1: GEMM (500->16) then aggregate in 16-dim ----
  gemm_xw1_kernel<<<cdiv((long long)mTiles * 32, T), blk, 0, stream>>>(X, W1, bufA, N);
  init_agg_kernel<<<cdiv(n16, T), blk, 0, stream>>>(bufA, bufB, dis, n16);
  edge_agg_kernel<<<cdiv((long long)E * HID, T), blk, 0, stream>>>(bufA, bufB, dis, src, dst, E);
  bias_relu_kernel<<<cdiv(n16, T), blk, 0, stream>>>(bufB, b1, bufA, n16);

  // ---- layer 2: GEMM (16->16) then aggregate ----
  gemm_k16_kernel<<<cdiv((long long)mTiles * 32, T), blk, 0, stream>>>(bufA, W2, (const float*)nullptr, bufB, N, HID);
  init_agg_kernel<<<cdiv(n16, T), blk, 0, stream>>>(bufB, bufA, dis, n16);
  edge_agg_kernel<<<cdiv((long long)E * HID, T), blk, 0, stream>>>(bufB, bufA, dis, src, dst, E);
  bias_relu_kernel<<<cdiv(n16, T), blk, 0, stream>>>(bufA, b2, bufB, n16);

  // ---- layer 3: aggregate FIRST (16-dim), then GEMM (16->500) + b3 ----
  init_agg_kernel<<<cdiv(n16, T), blk, 0, stream>>>(bufB, bufA, dis, n16);
  edge_agg_kernel<<<cdiv((long long)E * HID, T), blk, 0, stream>>>(bufB, bufA, dis, src, dst, E);
  const int nTiles3 = (OUTD + 15) >> 4;           // 32
  gemm_k16_kernel<<<cdiv((long long)mTiles * nTiles3 * 32, T), blk, 0, stream>>>(bufA, W3, b3, out, N, OUTD);
}